// SageConvolution_29549374996932
// MI455X (gfx1250) — compile-verified
//
#include <hip/hip_runtime.h>

typedef __attribute__((ext_vector_type(2))) float v2f;
typedef __attribute__((ext_vector_type(8))) float v8f;
typedef __attribute__((ext_vector_type(4))) int   v4i;

#define D_FEAT 128
#define MTILE  64   // rows of x per block (4 x 16-row WMMA sub-tiles)

#if __has_builtin(__builtin_amdgcn_global_load_async_to_lds_b128)
#define HAVE_ASYNC_LDS 1
#else
#define HAVE_ASYNC_LDS 0
#endif

// ---------------------------------------------------------------------------
// Fused dual GEMM:  h_neigh = x @ Wl   and   out = x @ Wr + bias
// Block = 256 threads = 8 wave32.  M-tile = 64 rows staged in 32 KB LDS
// via the async global->LDS path (ASYNCcnt).  Wave w owns output columns
// [16w,16w+16); each B fragment (4 global b32 loads) feeds 8
// V_WMMA_F32_16X16X4_F32 ops (4 M sub-tiles x 2 weight matrices).
// ---------------------------------------------------------------------------
__global__ __launch_bounds__(256)
void sage_gemm_kernel(const float* __restrict__ x,
                      const float* __restrict__ Wl,
                      const float* __restrict__ Wr,
                      const float* __restrict__ bias,
                      float* __restrict__ h_neigh,
                      float* __restrict__ out,
                      int n_nodes) {
    __shared__ float xs[MTILE * D_FEAT];       // 32 KB x-tile

    const int tid  = threadIdx.x;
    const int wave = tid >> 5;
    const int lane = tid & 31;
    const int row0 = blockIdx.x * MTILE;

    // Stage 64x128 fp32 x-tile: 2048 float4, 8 per thread (clamped tail).
    float4* xs4 = (float4*)xs;
    #pragma unroll
    for (int i = tid; i < MTILE * 32; i += 256) {
        const int r  = i >> 5;                 // 32 float4 per row
        const int cc = i & 31;
        int gr = row0 + r;
        if (gr >= n_nodes) gr = n_nodes - 1;   // clamp (tail block only)
        const float4* gsrc = (const float4*)(x + (size_t)gr * D_FEAT) + cc;
#if HAVE_ASYNC_LDS
        __builtin_amdgcn_global_load_async_to_lds_b128(
            (__attribute__((address_space(1))) v4i*)(void*)(uintptr_t)gsrc,
            (__attribute__((address_space(3))) v4i*)&xs4[i],
            0, 0);
#else
        xs4[i] = *gsrc;
#endif
    }
#if HAVE_ASYNC_LDS
#if __has_builtin(__builtin_amdgcn_s_wait_asynccnt)
    __builtin_amdgcn_s_wait_asynccnt(0);
#else
    asm volatile("s_wait_asynccnt 0" ::: "memory");
#endif
#endif
    __syncthreads();

    const int n0    = wave << 4;               // this wave's N-tile base
    const int nl    = lane & 15;               // N (for B/C/D) and M (for A)
    const int khalf = (lane >> 4) << 1;        // K split across lane halves

    v8f accL[4] = {{}, {}, {}, {}};            // x @ Wl tiles
    v8f accR[4] = {{}, {}, {}, {}};            // x @ Wr tiles

    const float* WlB = Wl + n0 + nl;           // W is [K=128][N=128]
    const float* WrB = Wr + n0 + nl;

    for (int k = 0; k < D_FEAT; k += 4) {
        const int ka = k + khalf;
        // B frags (4x16 f32): lane = N, VGPR v holds K = khalf + v
        v2f bl, br;
        bl.x = WlB[(size_t)(ka)     * D_FEAT];
        bl.y = WlB[(size_t)(ka + 1) * D_FEAT];
        br.x = WrB[(size_t)(ka)     * D_FEAT];
        br.y = WrB[(size_t)(ka + 1) * D_FEAT];

        #pragma unroll
        for (int ms = 0; ms < 4; ++ms) {
            // A frag (16x4 f32): lane = M, VGPR v holds K = khalf + v
            v2f a;
            a.x = xs[(ms * 16 + nl) * D_FEAT + ka];
            a.y = xs[(ms * 16 + nl) * D_FEAT + ka + 1];
            accL[ms] = __builtin_amdgcn_wmma_f32_16x16x4_f32(
                           false, a, false, bl, (short)0, accL[ms], false, false);
            accR[ms] = __builtin_amdgcn_wmma_f32_16x16x4_f32(
                           false, a, false, br, (short)0, accR[ms], false, false);
        }
    }

    const float bv = bias[n0 + nl];            // bias depends only on N

    // C/D layout: VGPR i, lane l -> (M = i + 8*(l/16), N = l%16)
    const int mhalf = (lane >> 4) << 3;
    if (row0 + MTILE <= n_nodes) {
        // Fast path (all blocks except the single tail block): straight-line
        // stores, no per-element exec masking.
        #pragma unroll
        for (int ms = 0; ms < 4; ++ms) {
            const size_t base = (size_t)(row0 + ms * 16 + mhalf) * D_FEAT + n0 + nl;
            #pragma unroll
            for (int i = 0; i < 8; ++i) {
                const size_t off = base + (size_t)i * D_FEAT;
                h_neigh[off] = accL[ms][i];
                out[off]     = accR[ms][i] + bv;   // initializes out pre-scatter
            }
        }
    } else {
        // Tail block: per-row guard.
        #pragma unroll
        for (int ms = 0; ms < 4; ++ms) {
            const int mbase = row0 + ms * 16 + mhalf;
            #pragma unroll
            for (int i = 0; i < 8; ++i) {
                const int m = mbase + i;
                if (m < n_nodes) {
                    const size_t off = (size_t)m * D_FEAT + n0 + nl;
                    h_neigh[off] = accL[ms][i];
                    out[off]     = accR[ms][i] + bv;
                }
            }
        }
    }
}

// ---------------------------------------------------------------------------
// Edge scatter: out[row] += val * h_neigh[col]   (one wave32 per edge;
// lane l owns 4 consecutive features -> float4 gather + 4 f32 atomics).
// h_neigh (51 MB) and out (51 MB) are L2-resident (192 MB L2).
// ---------------------------------------------------------------------------
__global__ __launch_bounds__(256)
void sage_edge_kernel(const int* __restrict__ erow,
                      const int* __restrict__ ecol,
                      const float* __restrict__ evalv,
                      const float* __restrict__ h,
                      float* __restrict__ out,
                      int n_edges) {
    const int lane   = threadIdx.x & 31;
    const int wid    = (int)((blockIdx.x * blockDim.x + threadIdx.x) >> 5);
    const int nwaves = (int)((gridDim.x * blockDim.x) >> 5);

    for (int e = wid; e < n_edges; e += nwaves) {
        const int   r = erow[e];               // wave-uniform broadcast loads
        const int   c = ecol[e];
        const float v = evalv[e];

        const float4 hv = ((const float4*)(h + (size_t)c * D_FEAT))[lane];
        float* op = out + (size_t)r * D_FEAT + (lane << 2);
        atomicAdd(op + 0, v * hv.x);
        atomicAdd(op + 1, v * hv.y);
        atomicAdd(op + 2, v * hv.z);
        atomicAdd(op + 3, v * hv.w);
    }
}

// ---------------------------------------------------------------------------
// inputs: x, edge_row, edge_col, edge_val, weight_l, weight_r, bias
// ---------------------------------------------------------------------------
extern "C" void kernel_launch(void* const* d_in, const int* in_sizes, int n_in,
                              void* d_out, int out_size, void* d_ws, size_t ws_size,
                              hipStream_t stream) {
    const float* x     = (const float*)d_in[0];
    const int*   erow  = (const int*)  d_in[1];
    const int*   ecol  = (const int*)  d_in[2];
    const float* eval_ = (const float*)d_in[3];
    const float* Wl    = (const float*)d_in[4];
    const float* Wr    = (const float*)d_in[5];
    const float* bias  = (const float*)d_in[6];

    const int n_nodes = in_sizes[0] / D_FEAT;
    const int n_edges = in_sizes[1];

    float* out     = (float*)d_out;
    float* h_neigh = (float*)d_ws;             // N*128*4 = 51.2 MB scratch

    // Phase 1: fused dual WMMA GEMM (also initializes d_out with x@Wr + bias).
    const int gemm_blocks = (n_nodes + MTILE - 1) / MTILE;
    sage_gemm_kernel<<<gemm_blocks, 256, 0, stream>>>(
        x, Wl, Wr, bias, h_neigh, out, n_nodes);

    // Phase 2: per-edge gather/scale/atomic-scatter (1 wave per edge).
    const int edge_blocks = (n_edges + 7) / 8; // 8 waves per 256-thread block
    sage_edge_kernel<<<edge_blocks, 256, 0, stream>>>(
        erow, ecol, eval_, h_neigh, out, n_edges);
}